// WHVILinear_46196668235888
// MI455X (gfx1250) — compile-verified
//
#include <hip/hip_runtime.h>
#include <hip/hip_bf16.h>
#include <stdint.h>
#include <math.h>

typedef __attribute__((ext_vector_type(2))) float v2f;
typedef __attribute__((ext_vector_type(4))) float v4f;
typedef __attribute__((ext_vector_type(8))) float v8f;

// async b128 copy operand types: toolchain expects 'int __vector(4)' pointers
// in explicit address spaces (AS1 = global, AS3 = LDS).
typedef __attribute__((vector_size(16))) int b128_t;
typedef __attribute__((address_space(1))) b128_t* gas_b128_ptr;
typedef __attribute__((address_space(3))) b128_t* lds_b128_ptr;

#define D_DIM        4096
#define N_ROWS       16384
#define ROWS_PER_BLK 16
#define LDS_STRIDE   4100   /* 4096 + 4 floats pad: row stride not a multiple of 256B -> no bank conflicts */
#define NTHREADS     256
#define QUADS        (ROWS_PER_BLK * (D_DIM / 4))   /* float4 chunks per block tile */

#ifndef HAVE_ASYNC_LDS
#if defined(__has_builtin)
#if __has_builtin(__builtin_amdgcn_global_load_async_to_lds_b128) && \
    __has_builtin(__builtin_amdgcn_s_wait_asynccnt)
#define HAVE_ASYNC_LDS 1
#endif
#endif
#endif
#ifndef HAVE_ASYNC_LDS
#define HAVE_ASYNC_LDS 0
#endif

// Sylvester Hadamard: H[i][k] = (-1)^popcount(i & k)
__device__ __forceinline__ float hsign(int i, int k) {
    return (__popc(i & k) & 1) ? -1.0f : 1.0f;
}

// One radix-16 FWHT stage over the whole 16 x 4096 LDS tile.
// STAGE 0: contract c (stride 1,   batch dim = the 16 rows, via lane columns)
// STAGE 1: contract b (stride 16,  tiles indexed by (row, a))
// STAGE 2: contract a (stride 256, tiles indexed by (row, b))
// Each 16x16 transform: D = H16 x tile, as 4 chained v_wmma_f32_16x16x4_f32.
template <int STAGE>
__device__ __forceinline__ void fwht_stage(float* lds, int lane, int wave, const v2f* hA) {
    const int n    = lane & 15;          // B/D column index held by this lane
    const int koff = (lane >> 4) << 1;   // lanes 0-15: K rows 4q+0/4q+1 ; lanes 16-31: 4q+2/4q+3
    const int mofs = (lane >> 4) << 3;   // D rows: acc[g] is row g (lanes 0-15) or g+8 (lanes 16-31)
    for (int t = wave * 32; t < (wave + 1) * 32; ++t) {
        int base, ks, ns;
        if (STAGE == 0) { base = t * 16;                                 ks = 1;   ns = LDS_STRIDE; }
        if (STAGE == 1) { base = (t >> 4) * LDS_STRIDE + (t & 15) * 256; ks = 16;  ns = 1; }
        if (STAGE == 2) { base = (t >> 4) * LDS_STRIDE + (t & 15) * 16;  ks = 256; ns = 1; }
        const int a0 = base + n * ns;

        v2f b[4];
#pragma unroll
        for (int q = 0; q < 4; ++q) {
            const int k0 = 4 * q + koff;
            b[q].x = lds[a0 + (k0 + 0) * ks];
            b[q].y = lds[a0 + (k0 + 1) * ks];
        }

        v8f acc = {0.f, 0.f, 0.f, 0.f, 0.f, 0.f, 0.f, 0.f};
#pragma unroll
        for (int q = 0; q < 4; ++q) {
            // (neg_a, A, neg_b, B, c_mod, C, reuse_a, reuse_b)
            acc = __builtin_amdgcn_wmma_f32_16x16x4_f32(false, hA[q], false, b[q],
                                                        (short)0, acc, false, false);
        }

#pragma unroll
        for (int g = 0; g < 8; ++g)
            lds[base + (mofs + g) * ks + n * ns] = acc[g];
    }
}

// u[k] = g_mu[k] + softplus(g_rho[k]) * epsilon[k]
__global__ void whvi_prepare_u(const float* __restrict__ g_mu,
                               const float* __restrict__ g_rho,
                               const float* __restrict__ eps,
                               float* __restrict__ u) {
    const int i = blockIdx.x * blockDim.x + threadIdx.x;
    if (i < D_DIM) {
        const float rho = g_rho[i];
        const float sp  = (rho > 20.0f) ? rho : log1pf(expf(rho));
        u[i] = g_mu[i] + sp * eps[i];
    }
}

// out[n,:] = (s1/64) * H4096( u * H4096( s2 * x[n,:] ) )   (H applied unnormalized; 1/sqrt(D)=1/64 folded into s1)
__launch_bounds__(NTHREADS)
__global__ void whvi_main(const float* __restrict__ x,
                          const float* __restrict__ s1,
                          const float* __restrict__ s2,
                          const float* __restrict__ u,
                          float* __restrict__ out) {
    extern __shared__ float lds[];
    const int tid  = threadIdx.x;
    const int lane = tid & 31;
    const int wave = tid >> 5;
    const int row0 = blockIdx.x * ROWS_PER_BLK;

    // H16 as WMMA A-operand lane constants (ISA 16x4 fp32 A layout)
    const int i16  = lane & 15;
    const int koff = (lane >> 4) << 1;
    v2f hA[4];
#pragma unroll
    for (int q = 0; q < 4; ++q) {
        const int k0 = 4 * q + koff;
        hA[q].x = hsign(i16, k0);
        hA[q].y = hsign(i16, k0 + 1);
    }

#if HAVE_ASYNC_LDS
    // async global -> LDS fill (b128 per lane), then scale by s2 in place
    for (int v = tid; v < QUADS; v += NTHREADS) {
        const int r = v >> 10, col = (v & 1023) << 2;
        const float* src = x + (size_t)(row0 + r) * D_DIM + col;
        float*       dst = lds + r * LDS_STRIDE + col;
        __builtin_amdgcn_global_load_async_to_lds_b128(
            (gas_b128_ptr)(uintptr_t)src,
            (lds_b128_ptr)(uintptr_t)dst,
            0, 0);
    }
    __builtin_amdgcn_s_wait_asynccnt(0);
    __syncthreads();
    for (int v = tid; v < QUADS; v += NTHREADS) {
        const int r = v >> 10, col = (v & 1023) << 2;
        v4f t = *(v4f*)(lds + r * LDS_STRIDE + col);
        t *= *(const v4f*)(s2 + col);
        *(v4f*)(lds + r * LDS_STRIDE + col) = t;
    }
#else
    for (int v = tid; v < QUADS; v += NTHREADS) {
        const int r = v >> 10, col = (v & 1023) << 2;
        v4f t = *(const v4f*)(x + (size_t)(row0 + r) * D_DIM + col);
        t *= *(const v4f*)(s2 + col);
        *(v4f*)(lds + r * LDS_STRIDE + col) = t;
    }
#endif
    __syncthreads();

    // first FWHT
    fwht_stage<0>(lds, lane, wave, hA); __syncthreads();
    fwht_stage<1>(lds, lane, wave, hA); __syncthreads();
    fwht_stage<2>(lds, lane, wave, hA); __syncthreads();

    // elementwise * u
    for (int v = tid; v < QUADS; v += NTHREADS) {
        const int r = v >> 10, col = (v & 1023) << 2;
        v4f t = *(v4f*)(lds + r * LDS_STRIDE + col);
        t *= *(const v4f*)(u + col);
        *(v4f*)(lds + r * LDS_STRIDE + col) = t;
    }
    __syncthreads();

    // second FWHT
    fwht_stage<0>(lds, lane, wave, hA); __syncthreads();
    fwht_stage<1>(lds, lane, wave, hA); __syncthreads();
    fwht_stage<2>(lds, lane, wave, hA); __syncthreads();

    // scale by s1 * 1/sqrt(D) and stream out
    for (int v = tid; v < QUADS; v += NTHREADS) {
        const int r = v >> 10, col = (v & 1023) << 2;
        v4f t = *(v4f*)(lds + r * LDS_STRIDE + col);
        t *= *(const v4f*)(s1 + col) * 0.015625f;   // 1/64 = 1/sqrt(4096)
        *(v4f*)(out + (size_t)(row0 + r) * D_DIM + col) = t;
    }
}

extern "C" void kernel_launch(void* const* d_in, const int* in_sizes, int n_in,
                              void* d_out, int out_size, void* d_ws, size_t ws_size,
                              hipStream_t stream) {
    (void)in_sizes; (void)n_in; (void)out_size; (void)ws_size;
    const float* x     = (const float*)d_in[0];
    const float* s1    = (const float*)d_in[1];
    const float* s2    = (const float*)d_in[2];
    const float* g_mu  = (const float*)d_in[3];
    const float* g_rho = (const float*)d_in[4];
    const float* eps   = (const float*)d_in[5];
    // d_in[6] = H: unused — H4096 = H16 (x) H16 (x) H16 applied implicitly via WMMA stages.
    float* out = (float*)d_out;
    float* u   = (float*)d_ws;   // 4096 floats of scratch

    whvi_prepare_u<<<dim3(D_DIM / 256), dim3(256), 0, stream>>>(g_mu, g_rho, eps, u);

    const size_t ldsBytes = (size_t)ROWS_PER_BLK * LDS_STRIDE * sizeof(float); // 262,400 B < 320 KB
    whvi_main<<<dim3(N_ROWS / ROWS_PER_BLK), dim3(NTHREADS), ldsBytes, stream>>>(x, s1, s2, u, out);
}